// H3_37993280700546
// MI455X (gfx1250) — compile-verified
//
#include <hip/hip_runtime.h>
#include <hip/hip_bf16.h>
#include <math.h>

typedef __attribute__((ext_vector_type(2))) float v2f;
typedef __attribute__((ext_vector_type(4))) float v4f;
typedef __attribute__((ext_vector_type(8))) float v8f;

#define Hdim 1024
#define Ldim 2048
#define Bdim 4
#define N2   32
#define ROWP 36   // LDS row pitch (floats): 16B-aligned rows, staggers banks

static __device__ __forceinline__ v8f wmma4(v2f a, v2f b, v8f c) {
  // V_WMMA_F32_16X16X4_F32: D = A(16x4) x B(4x16) + C, exact f32
  return __builtin_amdgcn_wmma_f32_16x16x4_f32(false, a, false, b, (short)0, c,
                                               false, false);
}

// -------------------------------------------------------------------------
// Kernel A: up = W_in(3H,H) @ u(B,H,L) + b_in ; u2' = u2*u1 ; stash u2', u3
// Each wave: 3 fused row-tiles (o0, o0+H, o0+2H) x 2 col-tiles (l0, l0+16).
// -------------------------------------------------------------------------
__global__ __launch_bounds__(128) void h3_gemm_in(
    const float* __restrict__ W, const float* __restrict__ U,
    const float* __restrict__ bias,
    float* __restrict__ u2p, float* __restrict__ u3) {
  const int lane = threadIdx.x & 31;
  const int wid  = blockIdx.x * 4 + (threadIdx.x >> 5);   // 16384 waves
  const int rg   = wid >> 8;          // 0..63  row group (o0 = rg*16)
  const int ct   = wid & 255;         // 0..255 column tile group
  const int b    = ct >> 6;           // 0..3
  const int l0   = (ct & 63) << 5;    // 0..2047 step 32
  const int o0   = rg << 4;

  const int m  = lane & 15;           // A fragment row
  const int kk = (lane >> 4) << 1;    // K sub-offset: lanes>=16 hold K=2,3
  const int n  = lane & 15;           // B fragment col

  const float* w1 = W + (size_t)(o0 + m) * Hdim + kk;
  const float* w2 = w1 + (size_t)Hdim * Hdim;
  const float* w3 = w2 + (size_t)Hdim * Hdim;
  const float* ub = U + (size_t)b * Hdim * Ldim + l0 + n;

  v8f c10 = {}, c11 = {}, c20 = {}, c21 = {}, c30 = {}, c31 = {};
  for (int h = 0; h < Hdim; h += 4) {
    const float* ur = ub + (size_t)(h + kk) * Ldim;
    v2f a1 = *(const v2f*)(w1 + h);
    v2f a2 = *(const v2f*)(w2 + h);
    v2f a3 = *(const v2f*)(w3 + h);
    v2f b0, b1;
    b0.x = ur[0];    b0.y = ur[Ldim];
    b1.x = ur[16];   b1.y = ur[Ldim + 16];
    __builtin_prefetch(ur + 4 * Ldim, 0, 3);   // WGP-scope prefetch (fills L0/L1)
    c10 = wmma4(a1, b0, c10);
    c11 = wmma4(a1, b1, c11);
    c20 = wmma4(a2, b0, c20);
    c21 = wmma4(a2, b1, c21);
    c30 = wmma4(a3, b0, c30);
    c31 = wmma4(a3, b1, c31);
  }

  const int hi8 = (lane >> 4) << 3;   // D-layout: lanes>=16 hold rows M=r+8
  float* p2 = u2p + ((size_t)b * Hdim + o0 + hi8) * Ldim + l0 + n;
  float* p3 = u3  + ((size_t)b * Hdim + o0 + hi8) * Ldim + l0 + n;
#pragma unroll
  for (int r = 0; r < 8; ++r) {
    const int row  = o0 + hi8 + r;
    const float b1v = bias[row];
    const float b2v = bias[row + Hdim];
    const float b3v = bias[row + 2 * Hdim];
    const size_t off = (size_t)r * Ldim;
    float x1 = c10[r] + b1v, x2 = c20[r] + b2v, x3 = c30[r] + b3v;
    p2[off]      = x2 * x1;            // u2' = u2 * u1
    p3[off]      = x3;                 // u3
    float y1 = c11[r] + b1v, y2 = c21[r] + b2v, y3 = c31[r] + b3v;
    p2[off + 16] = y2 * y1;
    p3[off + 16] = y3;
  }
}

// -------------------------------------------------------------------------
// Kernel B: S4D SSM as a diagonal recurrence. One wave per (b,h), lane=state n.
//   s_n[l] = w_n*s_n[l-1] + u2'[l]
//   y[l]   = 2*Re(sum_n Cd_n*s_n[l]) + D_h*u2'[l]
//   u3    <- u3 * y   (in place)
// Reduction over n done via LDS transpose once per 32-step chunk instead of a
// 5-step shuffle butterfly per step: ~2x fewer issue slots in the serial loop.
// -------------------------------------------------------------------------
__global__ __launch_bounds__(256) void h3_ssm_scan(
    const float* __restrict__ log_dt, const float* __restrict__ A_re,
    const float* __restrict__ A_im,  const float* __restrict__ C_re,
    const float* __restrict__ C_im,  const float* __restrict__ Dvec,
    const float* __restrict__ u2p,   float* __restrict__ u3) {
  __shared__ float xp[8][32 * ROWP];            // per-wave 32x32 transpose pad
  const int lane  = threadIdx.x & 31;
  const int wslot = threadIdx.x >> 5;
  const int wid   = blockIdx.x * 8 + wslot;     // 4096 waves
  const int b     = wid >> 10;
  const int h     = wid & 1023;
  const int idx   = h * N2 + lane;
  float* const tp = &xp[wslot][0];

  // ZOH discretization per (h, n)
  const float dt = __expf(log_dt[h]);
  const float Ar = -A_re[idx];          // A = -A_re + i*A_im
  const float Ai =  A_im[idx];
  const float er = Ar * dt, ei = Ai * dt;
  const float mag = __expf(er);
  const float wr = mag * __cosf(ei);    // w = exp(dt*A)
  const float wi = mag * __sinf(ei);
  const float cr = C_re[idx], ci = C_im[idx];
  const float nr = wr - 1.0f, ni = wi;  // (w - 1)
  const float tr = cr * nr - ci * ni;   // C*(w-1)
  const float ti = cr * ni + ci * nr;
  const float inv = 1.0f / (Ar * Ar + Ai * Ai);
  const float cdr = (tr * Ar + ti * Ai) * inv;   // Cd = C*(w-1)/A
  const float cdi = (ti * Ar - tr * Ai) * inv;
  const float Dh  = Dvec[h];

  const size_t base = ((size_t)b * Hdim + h) * Ldim;
  float sr = 0.0f, si = 0.0f;
  for (int l0 = 0; l0 < Ldim; l0 += 32) {
    const float uc = u2p[base + l0 + lane];   // lane t holds u[l0+t]
    // ---- serial phase: lane n advances state n, logs contribution per step
#pragma unroll
    for (int j = 0; j < 32; ++j) {
      const float uj = __shfl(uc, j, 32);     // broadcast timestep j
      const float srn = wr * sr - wi * si + uj;
      const float sin_ = wr * si + wi * sr;
      sr = srn; si = sin_;
      tp[j * ROWP + lane] = cdr * sr - cdi * si;   // c[n] at step j
    }
    // same-wave LDS RAW: DS ops are in-order, but fence explicitly (CDNA5
    // split counter) and stop the compiler moving loads above the stores.
    asm volatile("s_wait_dscnt 0x0" ::: "memory");
    // ---- reduce phase: lane t sums its own 32-wide row (8x ds_load_b128)
    const float* myrow = tp + lane * ROWP;
    v4f acc0 = {0.f, 0.f, 0.f, 0.f};
    v4f acc1 = {0.f, 0.f, 0.f, 0.f};
#pragma unroll
    for (int q = 0; q < 32; q += 8) {
      acc0 += *(const v4f*)(myrow + q);
      acc1 += *(const v4f*)(myrow + q + 4);
    }
    v4f acc = acc0 + acc1;
    const float sum = (acc.x + acc.y) + (acc.z + acc.w);
    const float y   = 2.0f * sum + Dh * uc;   // uc == u[l0+lane] for lane t
    const float u3v = u3[base + l0 + lane];
    u3[base + l0 + lane] = u3v * y;           // u3' = u3 * y
  }
}

// -------------------------------------------------------------------------
// Kernel C: z = W_out(2H,H) @ u3'(B,H,L) + b_out ; out = a * sigmoid(g)
// Each wave: 2 fused row-tiles (value o0, gate o0+H) x 2 col-tiles.
// -------------------------------------------------------------------------
__global__ __launch_bounds__(128) void h3_gemm_out(
    const float* __restrict__ W, const float* __restrict__ U,
    const float* __restrict__ bias, float* __restrict__ out) {
  const int lane = threadIdx.x & 31;
  const int wid  = blockIdx.x * 4 + (threadIdx.x >> 5);
  const int rg   = wid >> 8;
  const int ct   = wid & 255;
  const int b    = ct >> 6;
  const int l0   = (ct & 63) << 5;
  const int o0   = rg << 4;

  const int m  = lane & 15;
  const int kk = (lane >> 4) << 1;
  const int n  = lane & 15;

  const float* wa = W + (size_t)(o0 + m) * Hdim + kk;
  const float* wg = wa + (size_t)Hdim * Hdim;
  const float* ub = U + (size_t)b * Hdim * Ldim + l0 + n;

  v8f ca0 = {}, ca1 = {}, cg0 = {}, cg1 = {};
  for (int h = 0; h < Hdim; h += 4) {
    const float* ur = ub + (size_t)(h + kk) * Ldim;
    v2f aa = *(const v2f*)(wa + h);
    v2f ag = *(const v2f*)(wg + h);
    v2f b0, b1;
    b0.x = ur[0];    b0.y = ur[Ldim];
    b1.x = ur[16];   b1.y = ur[Ldim + 16];
    __builtin_prefetch(ur + 4 * Ldim, 0, 3);
    ca0 = wmma4(aa, b0, ca0);
    ca1 = wmma4(aa, b1, ca1);
    cg0 = wmma4(ag, b0, cg0);
    cg1 = wmma4(ag, b1, cg1);
  }

  const int hi8 = (lane >> 4) << 3;
  float* po = out + ((size_t)b * Hdim + o0 + hi8) * Ldim + l0 + n;
#pragma unroll
  for (int r = 0; r < 8; ++r) {
    const int row = o0 + hi8 + r;
    const float ba = bias[row];
    const float bg = bias[row + Hdim];
    const size_t off = (size_t)r * Ldim;
    float a0 = ca0[r] + ba, g0 = cg0[r] + bg;
    po[off]      = a0 / (1.0f + __expf(-g0));  // GLU
    float a1 = ca1[r] + ba, g1 = cg1[r] + bg;
    po[off + 16] = a1 / (1.0f + __expf(-g1));
  }
}

extern "C" void kernel_launch(void* const* d_in, const int* in_sizes, int n_in,
                              void* d_out, int out_size, void* d_ws, size_t ws_size,
                              hipStream_t stream) {
  (void)in_sizes; (void)n_in; (void)out_size; (void)ws_size;
  const float* u      = (const float*)d_in[0];
  const float* W_in   = (const float*)d_in[1];
  const float* b_in   = (const float*)d_in[2];
  const float* log_dt = (const float*)d_in[3];
  const float* A_re   = (const float*)d_in[4];
  const float* A_im   = (const float*)d_in[5];
  const float* C_re   = (const float*)d_in[6];
  const float* C_im   = (const float*)d_in[7];
  const float* Dv     = (const float*)d_in[8];
  const float* W_out  = (const float*)d_in[9];
  const float* b_out  = (const float*)d_in[10];
  float* out = (float*)d_out;

  float* ws  = (float*)d_ws;
  float* u2p = ws;                                   // (B,H,L) 32 MB
  float* u3  = ws + (size_t)Bdim * Hdim * Ldim;      // (B,H,L) 32 MB

  // GEMM-in + gate fuse: 16384 waves = 4096 blocks x 4 waves
  h3_gemm_in<<<dim3(4096), dim3(128), 0, stream>>>(W_in, u, b_in, u2p, u3);
  // SSM recurrence: 4096 waves = 512 blocks x 8 waves
  h3_ssm_scan<<<dim3(512), dim3(256), 0, stream>>>(log_dt, A_re, A_im,
                                                   C_re, C_im, Dv, u2p, u3);
  // GEMM-out + GLU: 16384 waves
  h3_gemm_out<<<dim3(4096), dim3(128), 0, stream>>>(W_out, u3, b_out, out);
}